// HyperGCNBlock_27556510171434
// MI455X (gfx1250) — compile-verified
//
#include <hip/hip_runtime.h>
#include <hip/hip_bf16.h>

#define N_NODES 100000
#define N_EDGES 100000
#define NNZ     1600000
#define DIM     128
#define LN_EPS  1e-5f

typedef __attribute__((ext_vector_type(2))) float v2f;
typedef __attribute__((ext_vector_type(8))) float v8f;

// ---------------------------------------------------------------------------
// Kernel 1: xw = x @ W   via V_WMMA_F32_16X16X4_F32 (full fp32 precision).
// One 256-thread block (8 waves) per 16-row block of x. W (128x128) and the
// 16x128 A-slab are staged in LDS with padded stride 132 to avoid bank
// conflicts on the strided A-fragment reads. Each wave owns one 16-col tile.
// ---------------------------------------------------------------------------
#define LDS_STRIDE 132

__global__ __launch_bounds__(256) void gemm_xw_wmma(const float* __restrict__ x,
                                                    const float* __restrict__ w,
                                                    float* __restrict__ xw) {
    __shared__ float sW[DIM * LDS_STRIDE];   // 128 rows (K) x 128 cols (N), padded
    __shared__ float sA[16 * LDS_STRIDE];    // 16 rows (M) x 128 cols (K), padded

    const int tid = threadIdx.x;
    const int rowBase = blockIdx.x * 16;

    // Stage W: row k of W -> sW[k*LDS_STRIDE + n]
    for (int i = tid; i < DIM * DIM; i += 256) {
        int r = i >> 7, c = i & 127;
        sW[r * LDS_STRIDE + c] = w[i];
    }
    // Stage A block: row m of x -> sA[m*LDS_STRIDE + k]
    for (int i = tid; i < 16 * DIM; i += 256) {
        int r = i >> 7, c = i & 127;
        sA[r * LDS_STRIDE + c] = x[(size_t)(rowBase + r) * DIM + c];
    }
    __syncthreads();

    const int wave = tid >> 5;        // 0..7 -> N tile
    const int lane = tid & 31;
    const int half = lane >> 4;       // 0 or 1
    const int l15  = lane & 15;
    const int colBase = wave * 16;

    v8f acc = {};
#pragma unroll
    for (int k = 0; k < DIM; k += 4) {
        const int kk = k + 2 * half;
        // A fragment: M = l15, K = kk, kk+1 (contiguous -> ds_load_b64)
        v2f a = *(const v2f*)&sA[l15 * LDS_STRIDE + kk];
        // B fragment: N = colBase + l15, K = kk, kk+1
        v2f b;
        b.x = sW[(kk    ) * LDS_STRIDE + colBase + l15];
        b.y = sW[(kk + 1) * LDS_STRIDE + colBase + l15];
        acc = __builtin_amdgcn_wmma_f32_16x16x4_f32(
            /*neg_a=*/false, a, /*neg_b=*/false, b,
            /*c_mod=*/(short)0, acc, /*reuse_a=*/false, /*reuse_b=*/false);
    }

    // D layout: VGPR r holds row (r + 8*half), cols = l15 across lanes 0..15
#pragma unroll
    for (int r = 0; r < 8; ++r) {
        xw[(size_t)(rowBase + r + 8 * half) * DIM + colBase + l15] = acc[r];
    }
}

// ---------------------------------------------------------------------------
// Kernel 2: weighted node degree Dn and edge size Be.
// ---------------------------------------------------------------------------
__global__ __launch_bounds__(256) void degrees_kernel(const int* __restrict__ node_idx,
                                                      const int* __restrict__ edge_idx,
                                                      const float* __restrict__ he_w,
                                                      float* __restrict__ Dn,
                                                      float* __restrict__ Be) {
    int i = blockIdx.x * 256 + threadIdx.x;
    if (i >= NNZ) return;
    int n = node_idx[i];
    int e = edge_idx[i];
    atomicAdd(&Dn[n], he_w[e]);
    atomicAdd(&Be[e], 1.0f);
}

// ---------------------------------------------------------------------------
// Kernel 3: nodes -> hyperedges:  eacc[edge] += xw[node]   (one wave32 per
// incidence, 4 floats/lane).
// ---------------------------------------------------------------------------
__global__ __launch_bounds__(256) void scatter_n2e(const float* __restrict__ xw,
                                                   const int* __restrict__ node_idx,
                                                   const int* __restrict__ edge_idx,
                                                   float* __restrict__ eacc) {
    int inc = (blockIdx.x * 256 + threadIdx.x) >> 5;
    int lane = threadIdx.x & 31;
    if (inc >= NNZ) return;
    int n = node_idx[inc];
    int e = edge_idx[inc];
    const float* __restrict__ src = xw + (size_t)n * DIM;
    float* __restrict__ dst = eacc + (size_t)e * DIM;
#pragma unroll
    for (int j = 0; j < 4; ++j) {
        int d = lane + 32 * j;
        atomicAdd(&dst[d], src[d]);
    }
}

// ---------------------------------------------------------------------------
// Kernel 4: hyperedges -> nodes:  out[node] += w_inc * (eacc[edge] / Be[edge])
// Be^{-1} folded in on the fly (saves a full 51 MB pass over eacc).
// ---------------------------------------------------------------------------
__global__ __launch_bounds__(256) void scatter_e2n(const float* __restrict__ eacc,
                                                   const float* __restrict__ Be,
                                                   const float* __restrict__ he_w,
                                                   const int* __restrict__ node_idx,
                                                   const int* __restrict__ edge_idx,
                                                   float* __restrict__ out_acc) {
    int inc = (blockIdx.x * 256 + threadIdx.x) >> 5;
    int lane = threadIdx.x & 31;
    if (inc >= NNZ) return;
    int n = node_idx[inc];
    int e = edge_idx[inc];
    float be = Be[e];
    float s = (be > 0.0f ? 1.0f / be : 0.0f) * he_w[e];
    const float* __restrict__ src = eacc + (size_t)e * DIM;
    float* __restrict__ dst = out_acc + (size_t)n * DIM;
#pragma unroll
    for (int j = 0; j < 4; ++j) {
        int d = lane + 32 * j;
        atomicAdd(&dst[d], src[d] * s);
    }
}

// ---------------------------------------------------------------------------
// Kernel 5: finalize (in place on d_out):
//   conv = acc * Dn_inv + conv_b ; h = relu(x + conv) ; out = layernorm(h)
// One wave32 per node row (128 = 32 lanes x 4), shuffle reductions.
// ---------------------------------------------------------------------------
__global__ __launch_bounds__(256) void finalize_kernel(const float* __restrict__ x,
                                                       const float* __restrict__ Dn,
                                                       const float* __restrict__ conv_b,
                                                       const float* __restrict__ ln_g,
                                                       const float* __restrict__ ln_b,
                                                       float* __restrict__ out) {
    int row = (blockIdx.x * 256 + threadIdx.x) >> 5;
    int lane = threadIdx.x & 31;
    if (row >= N_NODES) return;

    float dn = Dn[row];
    float dninv = dn > 0.0f ? 1.0f / dn : 0.0f;

    float h[4];
    float sum = 0.0f;
#pragma unroll
    for (int j = 0; j < 4; ++j) {
        int d = lane + 32 * j;
        float c = out[(size_t)row * DIM + d] * dninv + conv_b[d];
        float v = x[(size_t)row * DIM + d] + c;
        v = v > 0.0f ? v : 0.0f;   // ReLU(residual + conv)
        h[j] = v;
        sum += v;
    }
#pragma unroll
    for (int off = 16; off > 0; off >>= 1) sum += __shfl_xor(sum, off, 32);
    float mu = sum * (1.0f / 128.0f);

    float vs = 0.0f;
#pragma unroll
    for (int j = 0; j < 4; ++j) {
        float dlt = h[j] - mu;
        vs += dlt * dlt;
    }
#pragma unroll
    for (int off = 16; off > 0; off >>= 1) vs += __shfl_xor(vs, off, 32);
    float rstd = rsqrtf(vs * (1.0f / 128.0f) + LN_EPS);

#pragma unroll
    for (int j = 0; j < 4; ++j) {
        int d = lane + 32 * j;
        out[(size_t)row * DIM + d] = (h[j] - mu) * rstd * ln_g[d] + ln_b[d];
    }
}

// ---------------------------------------------------------------------------
extern "C" void kernel_launch(void* const* d_in, const int* in_sizes, int n_in,
                              void* d_out, int out_size, void* d_ws, size_t ws_size,
                              hipStream_t stream) {
    const float* x      = (const float*)d_in[0];
    const int*   he_idx = (const int*)d_in[1];          // [2, NNZ]
    const float* he_w   = (const float*)d_in[2];        // [N_EDGES]
    const float* conv_w = (const float*)d_in[3];        // [128,128]
    const float* conv_b = (const float*)d_in[4];
    const float* ln_g   = (const float*)d_in[5];
    const float* ln_b   = (const float*)d_in[6];
    float* out = (float*)d_out;                         // [N_NODES,128]

    const int* node_idx = he_idx;                       // row 0
    const int* edge_idx = he_idx + NNZ;                 // row 1

    // Workspace layout (~103 MB total)
    float* xw   = (float*)d_ws;                         // N_NODES*DIM
    float* eacc = xw + (size_t)N_NODES * DIM;           // N_EDGES*DIM
    float* Dn   = eacc + (size_t)N_EDGES * DIM;         // N_NODES
    float* Be   = Dn + N_NODES;                         // N_EDGES

    // Zero accumulators (capture-safe memset nodes)
    hipMemsetAsync(eacc, 0, (size_t)N_EDGES * DIM * sizeof(float), stream);
    hipMemsetAsync(Dn,   0, (size_t)N_NODES * sizeof(float), stream);
    hipMemsetAsync(Be,   0, (size_t)N_EDGES * sizeof(float), stream);
    hipMemsetAsync(out,  0, (size_t)N_NODES * DIM * sizeof(float), stream);

    // 1) GEMM on the WMMA path
    gemm_xw_wmma<<<N_NODES / 16, 256, 0, stream>>>(x, conv_w, xw);

    // 2) Degrees
    degrees_kernel<<<(NNZ + 255) / 256, 256, 0, stream>>>(node_idx, edge_idx, he_w, Dn, Be);

    // 3) nodes -> edges (one wave per incidence)
    scatter_n2e<<<(NNZ * 32) / 256, 256, 0, stream>>>(xw, node_idx, edge_idx, eacc);

    // 4) edges -> nodes with Be^{-1} and w_inc folded in
    scatter_e2n<<<(NNZ * 32) / 256, 256, 0, stream>>>(eacc, Be, he_w, node_idx, edge_idx, out);

    // 5) bias + residual + ReLU + LayerNorm, in place on d_out
    finalize_kernel<<<(N_NODES * 32 + 255) / 256, 256, 0, stream>>>(x, Dn, conv_b, ln_g, ln_b, out);
}